// GraphConvGRU_25271587570213
// MI455X (gfx1250) — compile-verified
//
#include <hip/hip_runtime.h>
#include <hip/hip_bf16.h>

typedef __attribute__((ext_vector_type(2))) float v2f;
typedef __attribute__((ext_vector_type(8))) float v8f;

#define LOG2E 1.4426950408889634f

__device__ __forceinline__ float fexp2(float x) { return __builtin_amdgcn_exp2f(x); }
__device__ __forceinline__ float frcp(float x)  { return __builtin_amdgcn_rcpf(x); }
// sigmoid(x) = 1 / (1 + 2^(-x*log2e))   -> v_exp_f32 + v_rcp_f32
__device__ __forceinline__ float sigm(float x)  { return frcp(1.0f + fexp2(-LOG2E * x)); }
// tanh(x) = (1 - e^{-2x}) / (1 + e^{-2x})
__device__ __forceinline__ float tanh_f(float x) {
  float e = fexp2(-2.0f * LOG2E * x);
  return (1.0f - e) * frcp(1.0f + e);
}
// Force a wave-uniform load result into SGPRs (weights are shared by all lanes).
__device__ __forceinline__ float uload(const float* p) {
  return __int_as_float(__builtin_amdgcn_readfirstlane(__float_as_int(*p)));
}

// ---------------------------------------------------------------------------
// Kernel 1: time-invariant input projection  xp[b, 0:9] = x @ [Wr0;Wz0;Wh0]^T + b
// fp32 WMMA: D(16x16) = A(16x4) x B(4x16) + C, K looped 128/4 = 32 times.
// A layout: lane L (0-15): V0=x[M=L,k+0], V1=x[M=L,k+1]; lanes 16-31: K half 2/3.
// B layout: lane L: col N=L&15;  V0=B[k + khalf], V1=B[k + khalf + 1].
// D layout: lane L: N=L&15, M=(L>>4)*8 + vgpr.
// ---------------------------------------------------------------------------
__global__ __launch_bounds__(256) void gcgru_proj_wmma(
    const float* __restrict__ x,
    const float* __restrict__ Wr0, const float* __restrict__ br0,
    const float* __restrict__ Wz0, const float* __restrict__ bz0,
    const float* __restrict__ Wh0, const float* __restrict__ bh0,
    float* __restrict__ xp, int B)
{
  const int lane  = threadIdx.x & 31;
  const int wave  = threadIdx.x >> 5;
  const int tile  = blockIdx.x * (blockDim.x >> 5) + wave;
  const int b0    = tile << 4;
  if (b0 >= B) return;

  const int idx   = lane & 15;         // A: row M; B/D: col N
  const int khalf = (lane >> 4) << 1;  // 0 or 2

  // B operand: column n of (9x128 stacked weights)^T, i.e. row n of the weights.
  const float* wrow;
  float bias;
  if (idx < 3)      { wrow = Wr0 + (size_t)idx * 128;       bias = br0[idx];     }
  else if (idx < 6) { wrow = Wz0 + (size_t)(idx - 3) * 128; bias = bz0[idx - 3]; }
  else if (idx < 9) { wrow = Wh0 + (size_t)(idx - 6) * 128; bias = bh0[idx - 6]; }
  else              { wrow = Wr0;                           bias = 0.0f;         } // dead cols

  const float* arow = x + (size_t)(b0 + idx) * 128 + khalf;
  const float* brow = wrow + khalf;

  v8f acc;   // C layout: every VGPR of this lane shares the same column N -> bias[N]
#pragma unroll
  for (int i = 0; i < 8; ++i) acc[i] = bias;

#pragma unroll
  for (int k = 0; k < 128; k += 4) {
    v2f a = *(const v2f*)(arow + k);   // 8B aligned (khalf even, k%4==0)
    v2f b = *(const v2f*)(brow + k);
    acc = __builtin_amdgcn_wmma_f32_16x16x4_f32(
        /*neg_a=*/false, a, /*neg_b=*/false, b,
        /*c_mod=*/(short)0, acc, /*reuse_a=*/false, /*reuse_b=*/false);
  }

  if (idx < 9) {
    const int mbase = (lane >> 4) << 3;
#pragma unroll
    for (int r = 0; r < 8; ++r)
      xp[(size_t)(b0 + mbase + r) * 9 + idx] = acc[r];
  }
}

// ---------------------------------------------------------------------------
// Kernel 2: 100-step 2-layer GraphConv-GRU recurrence, one thread per batch
// element, all state in VGPRs, weights in SGPRs, graph hard-coded (clique-
// factored exact form of D^-1/2 A D^-1/2).
// ---------------------------------------------------------------------------
__global__ __launch_bounds__(256) void gcgru_steps(
    const float* __restrict__ xp,
    const float* __restrict__ Wg0, const float* __restrict__ bg0,
    const float* __restrict__ Wr1, const float* __restrict__ br1,
    const float* __restrict__ Wz1, const float* __restrict__ bz1,
    const float* __restrict__ Wh1, const float* __restrict__ bh1,
    const float* __restrict__ Wg1, const float* __restrict__ bg1,
    float* __restrict__ out, int B)
{
  const int b = blockIdx.x * blockDim.x + threadIdx.x;
  if (b >= B) return;

  constexpr int NN = 22, HH = 3, TT = 100;
  constexpr float D13 = 0.2773500981126146f;  // 1/sqrt(13): hub nodes 0, 9
  constexpr float D5  = 0.4472135954999579f;  // 1/sqrt(5):  clique-2 members
  const float DINV[NN] = { D13, .5f, .5f, D5, .5f, .5f, D5, .5f, .5f, D13, .5f,
                           .5f, D5, .5f, .5f, D5, .5f, .5f, .5f, .5f, .5f, .5f };
  const int CLIQ[NN] = { 0, 1, 0, 2, 1, 0, 2, 1, 0, 0, 1,
                         0, 2, 4, 3, 2, 4, 3, 4, 3, 4, 3 };  // single clique (hubs unused)
  const int CMEMB[5][6] = { {0,2,5,8,11,0}, {0,1,4,7,10,0}, {0,3,6,9,12,15},
                            {9,14,17,19,21,9}, {9,13,16,18,20,9} };
  const int CLEN[5] = {5,5,6,5,5};

  // Shared weights -> SGPRs.
  float wg0[9], wr1[9], wz1[9], wh1[9], wg1[9];
  float vbg0[3], vbr1[3], vbz1[3], vbh1[3], vbg1[3];
#pragma unroll
  for (int i = 0; i < 9; ++i) {
    wg0[i] = uload(Wg0 + i); wr1[i] = uload(Wr1 + i); wz1[i] = uload(Wz1 + i);
    wh1[i] = uload(Wh1 + i); wg1[i] = uload(Wg1 + i);
  }
#pragma unroll
  for (int i = 0; i < 3; ++i) {
    vbg0[i] = uload(bg0 + i); vbr1[i] = uload(br1 + i); vbz1[i] = uload(bz1 + i);
    vbh1[i] = uload(bh1 + i); vbg1[i] = uload(bg1 + i);
  }

  // Per-batch projected inputs (time invariant, broadcast over nodes).
  float xr[3], xz[3], xh[3];
  {
    const float* p = xp + (size_t)b * 9;
#pragma unroll
    for (int j = 0; j < 3; ++j) { xr[j] = p[j]; xz[j] = p[3 + j]; xh[j] = p[6 + j]; }
  }

  float h0[NN][HH], h1[NN][HH];
#pragma unroll
  for (int n = 0; n < NN; ++n)
#pragma unroll
    for (int j = 0; j < HH; ++j) { h0[n][j] = 0.f; h1[n][j] = 0.f; }

  float* optr = out + (size_t)b * (TT * NN * HH);

#pragma unroll 1
  for (int t = 0; t < TT; ++t, optr += NN * HH) {
    // Clique sums of dinv[n]*h[n] (old state) for both layers.
    float s0[5][HH], s1[5][HH];
#pragma unroll
    for (int c = 0; c < 5; ++c)
#pragma unroll
      for (int j = 0; j < HH; ++j) { s0[c][j] = 0.f; s1[c][j] = 0.f; }
#pragma unroll
    for (int c = 0; c < 5; ++c)
#pragma unroll
      for (int i = 0; i < 6; ++i)
        if (i < CLEN[c]) {
          const int n = CMEMB[c][i];
#pragma unroll
          for (int j = 0; j < HH; ++j) {
            s0[c][j] += DINV[n] * h0[n][j];
            s1[c][j] += DINV[n] * h1[n][j];
          }
        }

#pragma unroll
    for (int m = 0; m < NN; ++m) {
      float S0[HH], S1[HH];
#pragma unroll
      for (int j = 0; j < HH; ++j) {
        if (m == 0)      { S0[j] = s0[0][j] + s0[1][j] + s0[2][j];
                           S1[j] = s1[0][j] + s1[1][j] + s1[2][j]; }
        else if (m == 9) { S0[j] = s0[2][j] + s0[3][j] + s0[4][j];
                           S1[j] = s1[2][j] + s1[3][j] + s1[4][j]; }
        else             { S0[j] = s0[CLIQ[m]][j]; S1[j] = s1[CLIQ[m]][j]; }
      }
      const float dm = DINV[m];
      const float kd = ((m == 0) || (m == 9)) ? 3.0f * dm : dm;  // k_m * dinv[m]

      float ah0[HH], ah1[HH];
#pragma unroll
      for (int j = 0; j < HH; ++j) {
        ah0[j] = dm * (S0[j] - kd * h0[m][j]);   // (A_norm @ h0)[m]
        ah1[j] = dm * (S1[j] - kd * h1[m][j]);   // (A_norm @ h1)[m]
      }
      float hc0[HH], hc1[HH];
#pragma unroll
      for (int k = 0; k < HH; ++k) {
        hc0[k] = vbg0[k] + ah0[0]*wg0[k] + ah0[1]*wg0[3+k] + ah0[2]*wg0[6+k];
        hc1[k] = vbg1[k] + ah1[0]*wg1[k] + ah1[1]*wg1[3+k] + ah1[2]*wg1[6+k];
      }

      // Layer-0 GRU cell.
      float h0n[HH];
#pragma unroll
      for (int j = 0; j < HH; ++j) {
        float r  = sigm(xr[j] + hc0[j]);
        float z  = sigm(xz[j] + hc0[j]);
        float ht = tanh_f(xh[j] + r * hc0[j]);
        h0n[j] = (1.0f - z) * h0[m][j] + z * ht;
        h0[m][j] = h0n[j];
      }

      // Layer-1 GRU cell (input = h0n projected per node).
#pragma unroll
      for (int j = 0; j < HH; ++j) {
        float pr = vbr1[j] + h0n[0]*wr1[j*3+0] + h0n[1]*wr1[j*3+1] + h0n[2]*wr1[j*3+2];
        float pz = vbz1[j] + h0n[0]*wz1[j*3+0] + h0n[1]*wz1[j*3+1] + h0n[2]*wz1[j*3+2];
        float ph = vbh1[j] + h0n[0]*wh1[j*3+0] + h0n[1]*wh1[j*3+1] + h0n[2]*wh1[j*3+2];
        float r  = sigm(pr + hc1[j]);
        float z  = sigm(pz + hc1[j]);
        float ht = tanh_f(ph + r * hc1[j]);
        float h1n = (1.0f - z) * h1[m][j] + z * ht;
        h1[m][j] = h1n;
        __builtin_nontemporal_store(h1n, optr + m * HH + j);  // 216 MB: bypass-cache hint
      }
    }
  }
}

extern "C" void kernel_launch(void* const* d_in, const int* in_sizes, int n_in,
                              void* d_out, int out_size, void* d_ws, size_t ws_size,
                              hipStream_t stream) {
  (void)n_in; (void)out_size; (void)ws_size;
  const float* x   = (const float*)d_in[0];
  const float* Wr0 = (const float*)d_in[1];
  const float* br0 = (const float*)d_in[2];
  const float* Wz0 = (const float*)d_in[3];
  const float* bz0 = (const float*)d_in[4];
  const float* Wh0 = (const float*)d_in[5];
  const float* bh0 = (const float*)d_in[6];
  const float* Wg0 = (const float*)d_in[7];
  const float* bg0 = (const float*)d_in[8];
  const float* Wr1 = (const float*)d_in[9];
  const float* br1 = (const float*)d_in[10];
  const float* Wz1 = (const float*)d_in[11];
  const float* bz1 = (const float*)d_in[12];
  const float* Wh1 = (const float*)d_in[13];
  const float* bh1 = (const float*)d_in[14];
  const float* Wg1 = (const float*)d_in[15];
  const float* bg1 = (const float*)d_in[16];

  const int B = in_sizes[0] / 128;      // 8192
  float* xproj = (float*)d_ws;          // (B, 9) fp32 scratch

  const int tiles = (B + 15) / 16;      // one 16x16 WMMA tile per wave
  const int wavesPerBlock = 8;
  dim3 gridP((tiles + wavesPerBlock - 1) / wavesPerBlock), blockP(wavesPerBlock * 32);
  gcgru_proj_wmma<<<gridP, blockP, 0, stream>>>(x, Wr0, br0, Wz0, bz0, Wh0, bh0, xproj, B);

  dim3 gridS((B + 255) / 256), blockS(256);
  gcgru_steps<<<gridS, blockS, 0, stream>>>(xproj, Wg0, bg0, Wr1, br1, Wz1, bz1,
                                            Wh1, bh1, Wg1, bg1, (float*)d_out, B);
}